// SpatialTransformerAffine_13503377179119
// MI455X (gfx1250) — compile-verified
//
#include <hip/hip_runtime.h>
#include <math.h>

#define PI_F 3.14159265358979323846f

// Native Clang vector (not HIP_vector_type) so __builtin_nontemporal_store
// accepts it; lowers to a single global_store_b128 th:NT.
typedef float v4f __attribute__((ext_vector_type(4)));

// out[b,h,w] = bilinear(img[b], xs(w,h), ys(w,h)) with per-batch affine theta,
// reparameterized th = theta * M1 + M2 (see reference).
// Layout: img/out are [B,H,W] row-major (C==1).
//
// MI455X notes:
//  - Pure bandwidth/gather workload (~1 FLOP/byte): floor ~ 512MB / 23.3TB/s.
//  - Input gathers use default RT caching (L2-resident warped footprint).
//  - Output is streamed write-once: non-temporal stores keep it from evicting
//    the gather-reusable input lines in the 192MB L2 (CDNA5 TH=NT store hint).
//  - Affine map => next chunk's gather addresses are predictable: prefetch.
__global__ __launch_bounds__(256) void st_affine_bilinear_kernel(
    const float* __restrict__ img,    // [B,H,W]
    const float* __restrict__ theta,  // [B,2,3]
    float* __restrict__ out,          // [B,H,W]
    int H, int W)
{
    const int b = blockIdx.z;
    const int h = blockIdx.y;

    // ---- wave-uniform: reparameterize theta for this batch ----
    const float* t = theta + b * 6;
    const float th00 = fmaf(t[0], 1.0f,  0.5f);
    const float th01 = fmaf(t[1], PI_F, -0.5f * PI_F);
    const float th02 = fmaf(t[2], 0.2f, -0.1f);
    const float th10 = fmaf(t[3], PI_F, -0.5f * PI_F);
    const float th11 = fmaf(t[4], 1.0f,  0.5f);
    const float th12 = fmaf(t[5], 0.2f, -0.1f);

    const float invW = 1.0f / (float)(W - 1);
    const float invH = 1.0f / (float)(H - 1);
    const float sW   = (float)(W - 1);
    const float sH   = (float)(H - 1);
    const float yn   = (float)h * invH;

    // xs(w) = (th00*xn + cx) * (W-1);  ys(w) = (th10*xn + cy) * (H-1)
    const float cx = fmaf(th01, yn, th02);
    const float cy = fmaf(th11, yn, th12);

    const float* __restrict__ src = img + (size_t)b * (size_t)H * (size_t)W;
    float*       __restrict__ dst = out + ((size_t)b * (size_t)H + (size_t)h) * (size_t)W;

    constexpr int VEC = 4;
    const int span = blockDim.x * VEC;  // pixels covered per loop iteration

    // One bilinear sample at output column w (weights use CLIPPED integer
    // coords, matching reference edge semantics).
    auto sample = [&](int w) -> float {
        const float xn = (float)w * invW;
        const float xs = fmaf(th00, xn, cx) * sW;
        const float ys = fmaf(th10, xn, cy) * sH;

        int x0 = (int)floorf(xs), y0 = (int)floorf(ys);
        int x1 = x0 + 1,          y1 = y0 + 1;
        x0 = min(max(x0, 0), W - 1);
        x1 = min(max(x1, 0), W - 1);
        y0 = min(max(y0, 0), H - 1);
        y1 = min(max(y1, 0), H - 1);

        const float x0f = (float)x0, x1f = (float)x1;
        const float y0f = (float)y0, y1f = (float)y1;
        const float wxh = x1f - xs;   // toward x0 column
        const float wxl = xs - x0f;   // toward x1 column
        const float wyh = y1f - ys;   // toward y0 row
        const float wyl = ys - y0f;   // toward y1 row

        const float* __restrict__ row0 = src + (size_t)y0 * (size_t)W;
        const float* __restrict__ row1 = src + (size_t)y1 * (size_t)W;
        const float Ia = row0[x0];
        const float Ib = row1[x0];
        const float Ic = row0[x1];
        const float Id = row1[x1];

        const float colL = fmaf(wyh, Ia, wyl * Ib);  // x0 column
        const float colR = fmaf(wyh, Ic, wyl * Id);  // x1 column
        return fmaf(wxh, colL, wxl * colR);
    };

    // Prefetch predicted gather rows for this thread's NEXT chunk.
    auto prefetch_next = [&](int wn) {
        const float xnn = (float)wn * invW;
        const float xsn = fmaf(th00, xnn, cx) * sW;
        const float ysn = fmaf(th10, xnn, cy) * sH;
        int xp = (int)floorf(xsn);
        int yp = (int)floorf(ysn);
        xp = min(max(xp, 0), W - 1);
        const int yp0 = min(max(yp,     0), H - 1);
        const int yp1 = min(max(yp + 1, 0), H - 1);
        __builtin_prefetch(src + (size_t)yp0 * (size_t)W + (size_t)xp, 0, 3);
        __builtin_prefetch(src + (size_t)yp1 * (size_t)W + (size_t)xp, 0, 3);
    };

    if ((W & (VEC - 1)) == 0) {
        // Fast path: W multiple of 4; wave stores 512 contiguous bytes per
        // iteration via per-thread non-temporal global_store_b128.
        for (int w0 = threadIdx.x * VEC; w0 < W; w0 += span) {
            const int wn = w0 + span;
            if (wn < W) prefetch_next(wn);

            v4f v;
            v.x = sample(w0 + 0);
            v.y = sample(w0 + 1);
            v.z = sample(w0 + 2);
            v.w = sample(w0 + 3);
            __builtin_nontemporal_store(v, reinterpret_cast<v4f*>(dst + w0));
        }
    } else {
        // Generic path: scalar non-temporal stores.
        for (int w = threadIdx.x; w < W; w += blockDim.x) {
            __builtin_nontemporal_store(sample(w), dst + w);
        }
    }
}

extern "C" void kernel_launch(void* const* d_in, const int* in_sizes, int n_in,
                              void* d_out, int out_size, void* d_ws, size_t ws_size,
                              hipStream_t stream) {
    const float* img   = (const float*)d_in[0];  // [B,H,W,1] f32
    const float* theta = (const float*)d_in[1];  // [B,2,3,1] f32
    float*       out   = (float*)d_out;          // [B,H,W,1] f32
    (void)d_ws; (void)ws_size; (void)n_in; (void)out_size;

    // Recover dims deterministically from flat element counts:
    // theta has B*6 elements; fmap has B*H*W with H==W (reference setup).
    const int B  = in_sizes[1] / 6;
    const long long hw = (long long)in_sizes[0] / (long long)B;
    const int H  = (int)(sqrt((double)hw) + 0.5);
    const int W  = H;

    dim3 block(256, 1, 1);
    dim3 grid(1, (unsigned)H, (unsigned)B);  // one block per (batch, row); block loops over W
    st_affine_bilinear_kernel<<<grid, block, 0, stream>>>(img, theta, out, H, W);
}